// Node2Vec_27891517620414
// MI455X (gfx1250) — compile-verified
//
#include <hip/hip_runtime.h>
#include <hip/hip_bf16.h>

typedef float v2f __attribute__((ext_vector_type(2)));
typedef float v8f __attribute__((ext_vector_type(8)));

#define N_NODES 100000
#define N_EDGES 3200000
#define F_IN    256
#define R_DIM   16
#define D_DIM   128
#define B_BATCH 16384
#define LN_EPS  1e-5f

// ---------------------------------------------------------------------------
// init: deg[i] = 1 (self loop), stats = 0
// ---------------------------------------------------------------------------
__global__ void init_kernel(float* __restrict__ deg, float* __restrict__ stats) {
    int i = blockIdx.x * blockDim.x + threadIdx.x;
    if (i < N_NODES) deg[i] = 1.0f;
    if (i < 4) stats[i] = 0.0f;
}

// ---------------------------------------------------------------------------
// degree accumulation over edge destinations
// ---------------------------------------------------------------------------
__global__ void deg_kernel(const int* __restrict__ dst, float* __restrict__ deg) {
    int e = blockIdx.x * blockDim.x + threadIdx.x;
    if (e < N_EDGES) atomicAdd(&deg[dst[e]], 1.0f);
}

// deg -> rsqrt(deg) in place (deg >= 1 always, due to self loops)
__global__ void dinv_kernel(float* __restrict__ deg) {
    int i = blockIdx.x * blockDim.x + threadIdx.x;
    if (i < N_NODES) deg[i] = rsqrtf(deg[i]);
}

// ---------------------------------------------------------------------------
// GEMM1: h = x @ W^T via V_WMMA_F32_16X16X4_F32.
// One wave per 16-node tile. C[16,16] accumulated over K=256 in steps of 4.
// Epilogue also writes hagg = h * dinv^2 (the self-loop message).
// ---------------------------------------------------------------------------
__global__ void gemm1_kernel(const float* __restrict__ x,
                             const float* __restrict__ W,
                             const float* __restrict__ dinv,
                             float* __restrict__ h,
                             float* __restrict__ hagg) {
    const int lane = threadIdx.x;           // 0..31
    const int tile = blockIdx.x;            // 0..6249 (100000/16)
    const int m    = lane & 15;             // A row / B col for this lane
    const int kh   = (lane >> 4) << 1;      // K sub-offset: 0 or 2

    const long arow = (long)(tile * 16 + m) * F_IN;  // x row (A: m = lane&15)
    const long brow = (long)m * F_IN;                // W row (B col n = lane&15)

    v8f c = {0.f, 0.f, 0.f, 0.f, 0.f, 0.f, 0.f, 0.f};

#pragma unroll 8
    for (int k0 = 0; k0 < F_IN; k0 += 4) {
        // A frag: lane holds x[row m][k0+kh], x[row m][k0+kh+1]
        v2f a = *(const v2f*)(x + arow + k0 + kh);
        // B frag: B[k][n] = W[n][k] -> lane holds W[n][k0+kh], W[n][k0+kh+1]
        v2f b = *(const v2f*)(W + brow + k0 + kh);
        c = __builtin_amdgcn_wmma_f32_16x16x4_f32(
                /*neg_a=*/false, a, /*neg_b=*/false, b,
                /*c_mod=*/(short)0, c, /*reuse_a=*/false, /*reuse_b=*/false);
    }

    // C/D layout: n = lane&15 ; vgpr r holds m = r + 8*(lane>>4)
    const int nn = lane & 15;
    const int mh = (lane >> 4) << 3;
#pragma unroll
    for (int r = 0; r < 8; ++r) {
        const int  row = tile * 16 + mh + r;
        const long idx = (long)row * R_DIM + nn;
        const float v  = c[r];
        h[idx] = v;
        const float di = dinv[row];
        hagg[idx] = v * di * di;            // self-loop contribution
    }
}

// ---------------------------------------------------------------------------
// Edge scatter: 16 lanes per edge. hagg[dst] += h[src] * dinv[src]*dinv[dst]
// h / hagg are 6.4 MB each -> resident in 192 MB L2; atomics stay on-chip.
// ---------------------------------------------------------------------------
__global__ void scatter_kernel(const int* __restrict__ src,
                               const int* __restrict__ dst,
                               const float* __restrict__ h,
                               const float* __restrict__ dinv,
                               float* __restrict__ hagg) {
    const long t = (long)blockIdx.x * blockDim.x + threadIdx.x;
    const long e = t >> 4;
    const int  r = (int)(t & 15);
    if (e >= N_EDGES) return;
    const int s = src[e];
    const int d = dst[e];
    const float nrm = dinv[s] * dinv[d];
    atomicAdd(&hagg[(long)d * R_DIM + r], h[(long)s * R_DIM + r] * nrm);
}

// ---------------------------------------------------------------------------
// Global sum / sum-of-squares of (hagg + b) for graph-wide LayerNorm
// ---------------------------------------------------------------------------
__global__ void stats_kernel(const float* __restrict__ hagg,
                             const float* __restrict__ bvec,
                             float* __restrict__ stats) {
    __shared__ float s0[256];
    __shared__ float s1[256];
    float sum = 0.f, sq = 0.f;
    const long total = (long)N_NODES * R_DIM;
    for (long i = (long)blockIdx.x * blockDim.x + threadIdx.x; i < total;
         i += (long)gridDim.x * blockDim.x) {
        const float v = hagg[i] + bvec[i & 15];
        sum += v;
        sq  += v * v;
    }
    s0[threadIdx.x] = sum;
    s1[threadIdx.x] = sq;
    __syncthreads();
    for (int s = 128; s > 0; s >>= 1) {
        if ((int)threadIdx.x < s) {
            s0[threadIdx.x] += s0[threadIdx.x + s];
            s1[threadIdx.x] += s1[threadIdx.x + s];
        }
        __syncthreads();
    }
    if (threadIdx.x == 0) {
        atomicAdd(&stats[0], s0[0]);
        atomicAdd(&stats[1], s1[0]);
    }
}

__global__ void finalize_kernel(float* __restrict__ stats) {
    if (blockIdx.x == 0 && threadIdx.x == 0) {
        const float inv_n = 1.0f / (float)((long)N_NODES * R_DIM);
        const float mean  = stats[0] * inv_n;
        const float var   = stats[1] * inv_n - mean * mean;
        stats[2] = mean;
        stats[3] = rsqrtf(var + LN_EPS);
    }
}

// ---------------------------------------------------------------------------
// GEMM2 (batch rows only): gather hagg[batch] + b, LayerNorm affine, PReLU,
// then C[16,128] = A[16,16] @ trans[16,128] with f32 WMMA (4 K-steps, 8 N-tiles).
// One wave per 16 batch rows.
// ---------------------------------------------------------------------------
__global__ void gemm2_kernel(const float* __restrict__ hagg,
                             const float* __restrict__ trans,
                             const int* __restrict__ batch,
                             const float* __restrict__ bvec,
                             const float* __restrict__ ln_w,
                             const float* __restrict__ ln_b,
                             const float* __restrict__ prelu_a,
                             const float* __restrict__ stats,
                             float* __restrict__ out) {
    __shared__ float tileA[16 * 16];
    const int lane = threadIdx.x;
    const int tile = blockIdx.x;            // 0..1023 (16384/16)
    const float mean  = stats[2];
    const float istd  = stats[3];
    const float slope = prelu_a[0];

    if (lane < 16) {
        const int node = batch[tile * 16 + lane];
#pragma unroll
        for (int r = 0; r < R_DIM; ++r) {
            float v = hagg[(long)node * R_DIM + r] + bvec[r];
            v = (v - mean) * istd * ln_w[r] + ln_b[r];   // graph-wide LayerNorm
            v = (v >= 0.f) ? v : slope * v;              // PReLU
            tileA[lane * 16 + r] = v;
        }
    }
    __syncthreads();

    const int nn = lane & 15;
    const int kh = (lane >> 4) << 1;        // 0 or 2
    const int mh = (lane >> 4) << 3;        // 0 or 8

#pragma unroll
    for (int nt = 0; nt < 8; ++nt) {
        const int n0 = nt * 16;
        v8f c = {0.f, 0.f, 0.f, 0.f, 0.f, 0.f, 0.f, 0.f};
#pragma unroll
        for (int k0 = 0; k0 < R_DIM; k0 += 4) {
            const int kb = k0 + kh;
            v2f a;
            a.x = tileA[nn * 16 + kb];           // A[m][kb],   m = lane&15
            a.y = tileA[nn * 16 + kb + 1];       // A[m][kb+1]
            v2f b;
            b.x = trans[(long)kb * D_DIM + n0 + nn];       // B[kb][n]
            b.y = trans[(long)(kb + 1) * D_DIM + n0 + nn]; // B[kb+1][n]
            c = __builtin_amdgcn_wmma_f32_16x16x4_f32(
                    false, a, false, b, (short)0, c, false, false);
        }
#pragma unroll
        for (int r = 0; r < 8; ++r) {
            const int row = tile * 16 + mh + r;
            out[(long)row * D_DIM + n0 + nn] = c[r];
        }
    }
}

// ---------------------------------------------------------------------------
extern "C" void kernel_launch(void* const* d_in, const int* in_sizes, int n_in,
                              void* d_out, int out_size, void* d_ws, size_t ws_size,
                              hipStream_t stream) {
    const float* x       = (const float*)d_in[0];
    const int*   ei      = (const int*)  d_in[1];   // [2, E]: src row then dst row
    const float* trans   = (const float*)d_in[2];
    const int*   batch   = (const int*)  d_in[3];
    const float* W       = (const float*)d_in[4];
    const float* bvec    = (const float*)d_in[5];
    const float* ln_w    = (const float*)d_in[6];
    const float* ln_b    = (const float*)d_in[7];
    const float* prelu_a = (const float*)d_in[8];
    float* out = (float*)d_out;

    // workspace layout (fp32): h [N*16] | hagg [N*16] | deg/dinv [N] | stats [4]
    float* h     = (float*)d_ws;
    float* hagg  = h    + (size_t)N_NODES * R_DIM;
    float* deg   = hagg + (size_t)N_NODES * R_DIM;   // becomes dinv in place
    float* stats = deg  + (size_t)N_NODES;

    const int* srcp = ei;
    const int* dstp = ei + N_EDGES;

    init_kernel<<<(N_NODES + 255) / 256, 256, 0, stream>>>(deg, stats);
    deg_kernel<<<(N_EDGES + 255) / 256, 256, 0, stream>>>(dstp, deg);
    dinv_kernel<<<(N_NODES + 255) / 256, 256, 0, stream>>>(deg);

    gemm1_kernel<<<N_NODES / 16, 32, 0, stream>>>(x, W, deg, h, hagg);

    const long scatter_threads = (long)N_EDGES * 16;
    scatter_kernel<<<(int)((scatter_threads + 255) / 256), 256, 0, stream>>>(
        srcp, dstp, h, deg, hagg);

    stats_kernel<<<1024, 256, 0, stream>>>(hagg, bvec, stats);
    finalize_kernel<<<1, 32, 0, stream>>>(stats);

    gemm2_kernel<<<B_BATCH / 16, 32, 0, stream>>>(
        hagg, trans, batch, bvec, ln_w, ln_b, prelu_a, stats, out);
}